// FCNKANLayer_5299989643376
// MI455X (gfx1250) — compile-verified
//
#include <hip/hip_runtime.h>

typedef __attribute__((ext_vector_type(2))) float v2f;
typedef __attribute__((ext_vector_type(8))) float v8f;

#define B_SZ 256
#define I_SZ 128
#define O_SZ 128
#define H_SZ 32
// W1: [I,O,H,7]  W2: [I,O,1,H]  B1: [I,O,H]  B2: [I,O,1]

__global__ __launch_bounds__(256) void fcn_kan_wmma_kernel(
    const float* __restrict__ x,
    const float* __restrict__ W1,
    const float* __restrict__ W2,
    const float* __restrict__ B1,
    const float* __restrict__ B2,
    float* __restrict__ out)
{
  const int lane  = threadIdx.x & 31;
  const int wv    = threadIdx.x >> 5;
  const int gw    = blockIdx.x * 8 + wv;     // 2048 waves = 16 btiles x 128 o
  const int o     = gw & (O_SZ - 1);
  const int btile = gw >> 7;
  const int b0    = btile * 16;

  const int half = lane >> 4;                // lane group (0: lanes 0-15, 1: 16-31)
  const int n    = lane & 15;                // N column (B/C/D) or M row (A)
  const float padmul = half ? 0.f : 1.f;     // zero the f=7 pad slot

  v8f acc = {0.f, 0.f, 0.f, 0.f, 0.f, 0.f, 0.f, 0.f};
  float bacc = 0.f;                          // sum_i B2[i,o]

  for (int i = 0; i < I_SZ; ++i) {
    // ---- Fourier features: f = [x, sin x, sin 2x, sin 4x, cos x, cos 2x, cos 4x, 0]
    const float xv = x[(b0 + n) * I_SZ + i];
    const float s1 = __sinf(xv),        c1 = __cosf(xv);
    const float s2 = __sinf(2.0f * xv), c2 = __cosf(2.0f * xv);
    const float s4 = __sinf(4.0f * xv), c4 = __cosf(4.0f * xv);

    // A (16x4 f32) layout: VGPR0 = K(2*half), VGPR1 = K(2*half+1)
    v2f a0, a1;
    a0.x = half ? s2 : xv;   // f2 | f0
    a0.y = half ? s4 : s1;   // f3 | f1
    a1.x = half ? c4 : c1;   // f6 | f4
    a1.y = half ? 0.f : c2;  // f7(pad) | f5

    const long eb = (long)(i * O_SZ + o) * H_SZ;   // base into [I,O,H]
    bacc += B2[i * O_SZ + o];

#pragma unroll
    for (int hh = 0; hh < 2; ++hh) {
      const int  h     = hh * 16 + n;
      const long wbase = (eb + h) * 7;             // into W1 [.,.,h,7]

      // B (4x16 f32) layout mirrors A: VGPR0 = K(2*half), VGPR1 = K(2*half+1)
      v2f bA, bB;
      bA.x = W1[wbase + 2 * half];                 // f0 | f2
      bA.y = W1[wbase + 1 + 2 * half];             // f1 | f3
      bB.x = W1[wbase + 4 + 2 * half];             // f4 | f6
      bB.y = W1[wbase + 5] * padmul;               // f5 | f7(pad)

      // speculative prefetch of next-i W1 line (L2 warm-keeping)
      __builtin_prefetch(W1 + wbase + (long)O_SZ * H_SZ * 7, 0, 1);

      // h-tile [16b x 16h] = feat[16x8] @ W1tile[8x16], K=7 padded to 8
      v8f c = {0.f, 0.f, 0.f, 0.f, 0.f, 0.f, 0.f, 0.f};
      c = __builtin_amdgcn_wmma_f32_16x16x4_f32(false, a0, false, bA,
                                                (short)0, c, false, false);
      c = __builtin_amdgcn_wmma_f32_16x16x4_f32(false, a1, false, bB,
                                                (short)0, c, false, false);

      const float b1v = B1[eb + h];
      const float w2v = W2[eb + h];
#pragma unroll
      for (int r = 0; r < 8; ++r) {
        const float hv = c[r] + b1v;                                // pre-act
        const float sg = __builtin_amdgcn_rcpf(1.0f + __expf(-hv)); // sigmoid
        acc[r] += (hv * sg) * w2v;                                  // silu*W2
      }
    }
  }

  // one final butterfly reduction over the 16-lane N groups (sum over h),
  // then add the accumulated B2 bias.
#pragma unroll
  for (int r = 0; r < 8; ++r) {
    float v = acc[r];
    v += __shfl_xor(v, 1, 32);
    v += __shfl_xor(v, 2, 32);
    v += __shfl_xor(v, 4, 32);
    v += __shfl_xor(v, 8, 32);
    acc[r] = v + bacc;
  }

  // C/D layout: VGPR r holds row M=r (lanes 0-15) and M=r+8 (lanes 16-31)
  if (n == 0) {
#pragma unroll
    for (int r = 0; r < 8; ++r)
      out[(b0 + r + 8 * half) * O_SZ + o] = acc[r];
  }
}

extern "C" void kernel_launch(void* const* d_in, const int* in_sizes, int n_in,
                              void* d_out, int out_size, void* d_ws, size_t ws_size,
                              hipStream_t stream) {
  const float* x  = (const float*)d_in[0];
  const float* W1 = (const float*)d_in[1];
  const float* W2 = (const float*)d_in[2];
  const float* B1 = (const float*)d_in[3];
  const float* B2 = (const float*)d_in[4];
  float* out = (float*)d_out;

  // 2048 waves: 16 batch-tiles x 128 output columns; 8 wave32 per block
  fcn_kan_wmma_kernel<<<256, 256, 0, stream>>>(x, W1, W2, B1, B2, out);
}